// Scatt_TwoOrder_87978110091707
// MI455X (gfx1250) — compile-verified
//
#include <hip/hip_runtime.h>
#include <hip/hip_bf16.h>
#include <math.h>

typedef __attribute__((ext_vector_type(2))) float v2f;
typedef __attribute__((ext_vector_type(8))) float v8f;

#define TAPS 49          // 7x7
#define KPAD 56          // 7 filter rows padded to stride 8 -> 14 wmma steps of K=4
#define NSTEP 14
#define INW 40           // staged input tile row stride (37 used cols + pad, all written)

// ---------------------------------------------------------------------------
// Band-pass + modulus, WMMA-based.
//   in   : [B, G, H, W] fp32 planes
//   psi_r/psi_i : [24, 49] filters; plane g uses filter set (g%3)*8 + p
//   u_out: [B, G*8, H/2, W/2],  u[b][g*8+p] = sqrt(conv_r^2 + conv_i^2), stride 2, pad 3
//
// One wave: M=16 filters (8 real + 8 imag) x N=16 stride-2 pixels, K=49 taps.
// K is permuted so step t covers taps (ky=t>>1, kx=4*(t&1)+{0..3}); kx==7 pad
// columns hold zero weights in A, so B may read arbitrary finite LDS there.
// Per step: one ds_load_b64 for A, one for B (immediate offsets), one WMMA.
// Block = 8 waves covering 8 consecutive output rows, sharing a 21x40 tile.
// ---------------------------------------------------------------------------
__global__ __launch_bounds__(256)
void bandpass_mod_wmma(const float* __restrict__ in,
                       const float* __restrict__ psi_r,
                       const float* __restrict__ psi_i,
                       float* __restrict__ u_out,
                       int G, int H, int W)
{
    const int OW = W >> 1, OH = H >> 1;
    __shared__ float sA[16][KPAD];     // filter matrix, row stride 8 per tap-row
    __shared__ float sIn[21][INW];     // input tile rows 2*oy0-3 .. +17

    const int tid  = threadIdx.x;
    const int wave = tid >> 5;
    const int lane = tid & 31;

    const int bz   = blockIdx.z;       // b*G + g
    const int g    = bz % G;
    const int b    = bz / G;
    const int oy   = blockIdx.y * 8 + wave;
    const int ox0  = blockIdx.x * 16;
    const int fset = g % 3;

    // --- stage A into LDS: column j <-> tap (ky=j>>3, kx=j&7); kx==7 -> 0 ---
    for (int idx = tid; idx < 16 * KPAD; idx += 256) {
        int m  = idx / KPAD, j = idx % KPAD;
        int ky = j >> 3, kx = j & 7;
        float v = 0.f;
        if (kx < 7) {
            const float* wsrc = (m < 8) ? psi_r : psi_i;
            v = wsrc[(fset * 8 + (m & 7)) * TAPS + ky * 7 + kx];
        }
        sA[m][j] = v;
    }

    // --- stage input tile (zero at borders; all 21x40 entries written) ---
    const int iy0 = blockIdx.y * 16 - 3;   // = 2*(blockIdx.y*8) - 3
    const int ix0 = ox0 * 2 - 3;
    const float* plane = in + ((size_t)b * G + g) * (size_t)H * W;
    for (int idx = tid; idx < 21 * INW; idx += 256) {
        int r = idx / INW, c = idx % INW;
        int iy = iy0 + r, ix = ix0 + c;
        float v = 0.f;
        if (iy >= 0 && iy < H && ix >= 0 && ix < W)
            v = plane[(size_t)iy * W + ix];
        sIn[r][c] = v;
    }
    // prefetch next x-tile of this plane into cache (global_prefetch_b8)
    if (ix0 + 37 < W) {
        int py = iy0 < 0 ? 0 : iy0;
        __builtin_prefetch(plane + (size_t)py * W + ix0 + 37, 0, 1);
    }
    __syncthreads();

    // Fragment layout (ISA): A/B element at (row/col = lane%16, K = 2*(lane/16)+vgpr)
    const int mn   = lane & 15;
    const int half = lane >> 4;
    const float* sAf  = &sA[0][0];
    const float* sInf = &sIn[0][0];
    const int aOff = mn * KPAD + 2 * half;               // per-lane A base (elements)
    const int bOff = (2 * wave) * INW + 2 * mn + 2 * half; // per-lane B base

    v8f acc = {};
    #pragma unroll
    for (int t = 0; t < NSTEP; ++t) {
        v2f a  = *(const v2f*)(sAf  + aOff + 4 * t);
        v2f bb = *(const v2f*)(sInf + bOff + (t >> 1) * INW + (t & 1) * 4);
        acc = __builtin_amdgcn_wmma_f32_16x16x4_f32(
                  /*neg_a=*/false, a, /*neg_b=*/false, bb,
                  /*c_mod=*/(short)0, acc, /*reuse_a=*/false, /*reuse_b=*/false);
    }

    // --- modulus: C/D layout pairs filter p (lanes 0-15) with p+8 (lanes 16-31)
    // in the same VGPR -> combine via shfl_xor(16). ---
    float* uplane = u_out + ((size_t)b * G + g) * 8 * (size_t)OH * OW;
    #pragma unroll
    for (int r = 0; r < 8; ++r) {
        float v = acc[r];
        float o = __shfl_xor(v, 16, 32);
        float res = sqrtf(v * v + o * o);
        if (half == 0)
            uplane[((size_t)r * OH + oy) * OW + ox0 + mn] = res;
    }
}

// ---------------------------------------------------------------------------
// Depthwise 7x7, stride 1, pad 3, filter = phi[ch % 3]. LDS-tiled VALU.
//   in : [B, C, H, W];  out written at channel offset into [B, out_ch_tot, H, W]
// ---------------------------------------------------------------------------
__global__ __launch_bounds__(256)
void dw_phi_s1(const float* __restrict__ in, const float* __restrict__ phi,
               float* __restrict__ out, int C, int H, int W,
               int out_ch_off, int out_ch_tot)
{
    __shared__ float sIn[22][72];      // (16+6) x (64+6), padded
    __shared__ float sW[TAPS];
    const int tid = threadIdx.x;
    const int bz  = blockIdx.z;        // b*C + ch
    const int ch  = bz % C;
    const int b   = bz / C;
    const int y0  = blockIdx.y * 16;
    const int x0  = blockIdx.x * 64;
    const float* plane = in + (size_t)bz * H * W;

    if (tid < TAPS) sW[tid] = phi[(ch % 3) * TAPS + tid];
    for (int idx = tid; idx < 22 * 70; idx += 256) {
        int r = idx / 70, c = idx % 70;
        int iy = y0 - 3 + r, ix = x0 - 3 + c;
        sIn[r][c] = (iy >= 0 && iy < H && ix >= 0 && ix < W)
                        ? plane[(size_t)iy * W + ix] : 0.f;
    }
    __syncthreads();

    const int lx  = tid & 63;
    const int ly0 = tid >> 6;
    float* oplane = out + ((size_t)b * out_ch_tot + out_ch_off + ch) * (size_t)H * W;
    for (int yy = ly0; yy < 16; yy += 4) {
        float acc = 0.f;
        #pragma unroll
        for (int ky = 0; ky < 7; ++ky)
            #pragma unroll
            for (int kx = 0; kx < 7; ++kx)
                acc = fmaf(sW[ky * 7 + kx], sIn[yy + ky][lx + kx], acc);
        oplane[(size_t)(y0 + yy) * W + x0 + lx] = acc;
    }
}

// ---------------------------------------------------------------------------
// Depthwise 7x7, stride 2, pad 3, filter = phi[ch % 3]. LDS-tiled VALU.
//   in : [B, C, H, W];  out at channel offset into [B, out_ch_tot, H/2, W/2]
// ---------------------------------------------------------------------------
__global__ __launch_bounds__(256)
void dw_phi_s2(const float* __restrict__ in, const float* __restrict__ phi,
               float* __restrict__ out, int C, int H, int W,
               int out_ch_off, int out_ch_tot)
{
    const int OH = H >> 1, OW = W >> 1;
    __shared__ float sIn[21][72];      // rows 2*oy0-3..+17 (21), cols 2*ox0-3..+65 (69)
    __shared__ float sW[TAPS];
    const int tid = threadIdx.x;
    const int bz  = blockIdx.z;
    const int ch  = bz % C;
    const int b   = bz / C;
    const int oy0 = blockIdx.y * 8;
    const int ox0 = blockIdx.x * 32;
    const float* plane = in + (size_t)bz * H * W;

    if (tid < TAPS) sW[tid] = phi[(ch % 3) * TAPS + tid];
    for (int idx = tid; idx < 21 * 69; idx += 256) {
        int r = idx / 69, c = idx % 69;
        int iy = 2 * oy0 - 3 + r, ix = 2 * ox0 - 3 + c;
        sIn[r][c] = (iy >= 0 && iy < H && ix >= 0 && ix < W)
                        ? plane[(size_t)iy * W + ix] : 0.f;
    }
    __syncthreads();

    const int lx = tid & 31, ly = tid >> 5;   // 32 x 8 outputs
    float acc = 0.f;
    #pragma unroll
    for (int ky = 0; ky < 7; ++ky)
        #pragma unroll
        for (int kx = 0; kx < 7; ++kx)
            acc = fmaf(sW[ky * 7 + kx], sIn[2 * ly + ky][2 * lx + kx], acc);
    out[((size_t)b * out_ch_tot + out_ch_off + ch) * (size_t)OH * OW
        + (size_t)(oy0 + ly) * OW + ox0 + lx] = acc;
}

// ---------------------------------------------------------------------------
extern "C" void kernel_launch(void* const* d_in, const int* in_sizes, int n_in,
                              void* d_out, int out_size, void* d_ws, size_t ws_size,
                              hipStream_t stream) {
    const float* x     = (const float*)d_in[0];   // [32,3,512,512]
    const float* phi   = (const float*)d_in[1];   // [3,1,7,7]
    const float* psi_r = (const float*)d_in[2];   // [24,1,7,7]
    const float* psi_i = (const float*)d_in[3];   // [24,1,7,7]
    float* out = (float*)d_out;                   // [32,243,128,128]

    const int B = 32;
    // Workspace layout: s1 first; u1 and u2 share the region after it
    // (u1 is dead before u2 is written).  Peak = 227 MB + 453 MB.
    float* s1 = (float*)d_ws;                                   // [32,27,256,256]
    float* u1 = s1 + (size_t)B * 27 * 256 * 256;                // [32,24,256,256]
    float* u2 = u1;                                             // [32,216,128,128]

    // 1) s0_phi = phi * x, stride 2            -> s1 channels [0,3)
    dw_phi_s2<<<dim3(512/2/32, 512/2/8, B * 3), 256, 0, stream>>>(
        x, phi, s1, 3, 512, 512, 0, 27);

    // 2) u1 = |psi * x|, stride 2 (WMMA)       -> u1 [32,24,256,256]
    bandpass_mod_wmma<<<dim3(256/16, 256/8, B * 3), 256, 0, stream>>>(
        x, psi_r, psi_i, u1, 3, 512, 512);

    // 3) u1_smooth = phi[ch%3] * u1, stride 1  -> s1 channels [3,27)
    dw_phi_s1<<<dim3(256/64, 256/16, B * 24), 256, 0, stream>>>(
        u1, phi, s1, 24, 256, 256, 3, 27);

    // 4) s1_phi = phi[ch%3] * s1, stride 2     -> out channels [0,27)
    dw_phi_s2<<<dim3(256/2/32, 256/2/8, B * 27), 256, 0, stream>>>(
        s1, phi, out, 27, 256, 256, 0, 243);

    // 5) u2 = |psi[(g%3)*8+p] * s1|, stride 2 (WMMA) -> u2 [32,216,128,128]
    bandpass_mod_wmma<<<dim3(128/16, 128/8, B * 27), 256, 0, stream>>>(
        s1, psi_r, psi_i, u2, 27, 256, 256);

    // 6) u2_smooth = phi[ch%3] * u2, stride 1  -> out channels [27,243)
    dw_phi_s1<<<dim3(128/64, 128/16, B * 216), 256, 0, stream>>>(
        u2, phi, out, 216, 128, 128, 27, 243);
}